// PhobiaLoss_38972533244242
// MI455X (gfx1250) — compile-verified
//
#include <hip/hip_runtime.h>

// ---------------------------------------------------------------------------
// YOLO-style loss, MI455X (gfx1250). Pure streaming reduction:
//   ~321 MB read once -> 1 float. Bandwidth bound (~14 us floor @ 23.3 TB/s).
// Strategy: grid-stride over cells, non-temporal loads of only the used
// channels, per-lane f32 accumulators, WMMA (16x16x4 f32, A=ones) for the
// 32->16 lane fold, ds_swizzle (immediate xor patterns) for 16->1,
// deterministic 2-stage block reduction (no float atomics -> replay-stable).
// ---------------------------------------------------------------------------

typedef float v2f __attribute__((ext_vector_type(2)));
typedef float v4f __attribute__((ext_vector_type(4)));
typedef float v8f __attribute__((ext_vector_type(8)));

#define S_DIM   13
#define B_BOX   2
#define C_CLS   3
#define N_SAMP  16384
#define NCELLS  (N_SAMP * S_DIM * S_DIM)      // 2,768,896
#define PRED_C  (B_BOX * (5 + C_CLS))         // 16 floats / cell (64B, aligned)
#define TGT_C   (B_BOX * 5 + C_CLS)           // 13 floats / cell (4B aligned only)
#define TPB     256
#define NBLK_MAX 2048
#define WAVES_PER_BLK (TPB / 32)

__device__ __forceinline__ float softplus_f(float z) {
    // numerically stable, matches jax.nn.softplus
    return fmaxf(z, 0.0f) + log1pf(expf(-fabsf(z)));
}
__device__ __forceinline__ float sigmoid_f(float z) {
    return 1.0f / (1.0f + expf(-z));
}
__device__ __forceinline__ float sqrt_abs_f(float x) {
    return sqrtf(fabsf(x) + 1e-6f);
}

// ds_swizzle_b32 xor-swap with immediate group-of-32 pattern:
// offset[14:10]=xor_mask, offset[9:5]=or_mask(0), offset[4:0]=and_mask(0x1f)
// Pattern must be an integer constant expression -> template parameter.
template <int IMM>
__device__ __forceinline__ float swz_xor_add(float v) {
    int o = __builtin_amdgcn_ds_swizzle(__float_as_int(v), IMM);
    return v + __int_as_float(o);
}

// ---------------------------------------------------------------------------
// Full 32-lane sum using the matrix engine for the first fold.
// A = ones(16x4)  =>  D[m][n] = sum_k B[k][n].
// Per-lane value v goes in B's first VGPR, 0 in the second: the two B rows
// living in VGPR0 cover all 32 lanes exactly once, so each D column is the sum
// of two distinct lanes; summing the 16 columns (4 ds_swizzle xor steps)
// yields the full 32-lane sum regardless of the exact row->lane mapping.
// Exact f32 arithmetic throughout.
// ---------------------------------------------------------------------------
__device__ __forceinline__ float wave_reduce32(float v) {
    v2f a; a.x = 1.0f; a.y = 1.0f;   // A: all ones
    v2f b; b.x = v;    b.y = 0.0f;   // B: lane values in VGPR0, zeros in VGPR1
    v8f c = {};
    v8f d = __builtin_amdgcn_wmma_f32_16x16x4_f32(
        /*neg_a=*/false, a, /*neg_b=*/false, b,
        /*c_mod=*/(short)0, c, /*reuse_a=*/false, /*reuse_b=*/false);
    float h = d[0];                  // column sum for column (lane & 15)
    h = swz_xor_add<0x201F>(h);      // SWAPX8
    h = swz_xor_add<0x101F>(h);      // SWAPX4
    h = swz_xor_add<0x081F>(h);      // SWAPX2
    h = swz_xor_add<0x041F>(h);      // SWAPX1
    return h;                        // full 32-lane sum in every lane
}

__global__ __launch_bounds__(TPB)
void yolo_loss_main(const float* __restrict__ pred,
                    const float* __restrict__ tgt,
                    float* __restrict__ partials)
{
    float sObj = 0.0f, sNoobj = 0.0f, nObj = 0.0f, nNoobj = 0.0f;

    const int stride = (int)(gridDim.x * blockDim.x);
    for (int i = (int)(blockIdx.x * blockDim.x + threadIdx.x); i < NCELLS; i += stride) {
        const float* p = pred + (size_t)i * PRED_C;
        const float* t = tgt  + (size_t)i * TGT_C;

        // speculative prefetch of next grid-stride iteration
        __builtin_prefetch(pred + (size_t)(i + stride) * PRED_C, 0, 0);
        __builtin_prefetch(tgt  + (size_t)(i + stride) * TGT_C,  0, 0);

        // ---- predictions: only channels 0-4, 9, 10-12 are used ----
        v4f   p03   = __builtin_nontemporal_load((const v4f*)p);          // 16B aligned
        float p4    = __builtin_nontemporal_load(p + 4);
        float p9    = __builtin_nontemporal_load(p + 9);
        v2f   p1011 = __builtin_nontemporal_load((const v2f*)(p + 10));   // 8B aligned
        float p12   = __builtin_nontemporal_load(p + 12);

        // ---- targets: only channels 0-4, 10-12 are used (4B aligned) ----
        float t0  = __builtin_nontemporal_load(t + 0);
        float t1  = __builtin_nontemporal_load(t + 1);
        float t2  = __builtin_nontemporal_load(t + 2);
        float t3  = __builtin_nontemporal_load(t + 3);
        float t4  = __builtin_nontemporal_load(t + 4);
        float t10 = __builtin_nontemporal_load(t + 10);
        float t11 = __builtin_nontemporal_load(t + 11);
        float t12 = __builtin_nontemporal_load(t + 12);

        float objf   = (t4 > 0.0f) ? 1.0f : 0.0f;
        float noobjf = 1.0f - objf;

        // coord loss
        float px = sigmoid_f(p03.x);
        float py = sigmoid_f(p03.y);
        float dx = px - t0, dy = py - t1;
        float dw = sqrt_abs_f(p03.z) - sqrt_abs_f(t2);
        float dh = sqrt_abs_f(p03.w) - sqrt_abs_f(t3);
        float coord = dx * dx + dy * dy + dw * dw + dh * dh;

        // confidence losses (softplus(p4) shared between obj & noobj terms)
        float sp4 = softplus_f(p4);
        float bce_obj   = sp4 - t4 * p4;
        float noobj_bce = sp4 + softplus_f(p9);

        // class loss: cross-entropy from logits p10..p12, target = argmax(t10..t12)
        float la = p1011.x, lb = p1011.y, lc = p12;
        int   idx  = 0;
        float best = t10;
        if (t11 > best) { best = t11; idx = 1; }
        if (t12 > best) { best = t12; idx = 2; }
        float m  = fmaxf(la, fmaxf(lb, lc));
        float lz = m + logf(expf(la - m) + expf(lb - m) + expf(lc - m));
        float picked = (idx == 0) ? la : ((idx == 1) ? lb : lc);
        float cls = lz - picked;

        // fold static lambda weights for obj-gated terms: 5*coord + 1*bce + 1*cls
        sObj   += objf   * (5.0f * coord + bce_obj + cls);
        sNoobj += noobjf * noobj_bce;            // 0.5 weight applied at finalize
        nObj   += objf;
        nNoobj += noobjf;
    }

    // loop has reconverged: EXEC is all ones -> WMMA legal
    float r0 = wave_reduce32(sObj);
    float r1 = wave_reduce32(sNoobj);
    float r2 = wave_reduce32(nObj);
    float r3 = wave_reduce32(nNoobj);

    __shared__ float lds[WAVES_PER_BLK * 4];
    const int lane = threadIdx.x & 31;
    const int wv   = threadIdx.x >> 5;
    if (lane == 0) {
        lds[wv * 4 + 0] = r0;
        lds[wv * 4 + 1] = r1;
        lds[wv * 4 + 2] = r2;
        lds[wv * 4 + 3] = r3;
    }
    __syncthreads();
    if (threadIdx.x < 4) {
        float acc = 0.0f;
        for (int w = 0; w < WAVES_PER_BLK; ++w) acc += lds[w * 4 + threadIdx.x];
        partials[blockIdx.x * 4 + threadIdx.x] = acc;
    }
}

__global__ __launch_bounds__(TPB)
void yolo_loss_final(const float* __restrict__ partials, int nblk,
                     float* __restrict__ out)
{
    float a0 = 0.0f, a1 = 0.0f, a2 = 0.0f, a3 = 0.0f;
    for (int b = (int)threadIdx.x; b < nblk; b += TPB) {
        a0 += partials[b * 4 + 0];
        a1 += partials[b * 4 + 1];
        a2 += partials[b * 4 + 2];
        a3 += partials[b * 4 + 3];
    }

    float r0 = wave_reduce32(a0);
    float r1 = wave_reduce32(a1);
    float r2 = wave_reduce32(a2);
    float r3 = wave_reduce32(a3);

    __shared__ float lds[WAVES_PER_BLK * 4];
    const int lane = threadIdx.x & 31;
    const int wv   = threadIdx.x >> 5;
    if (lane == 0) {
        lds[wv * 4 + 0] = r0;
        lds[wv * 4 + 1] = r1;
        lds[wv * 4 + 2] = r2;
        lds[wv * 4 + 3] = r3;
    }
    __syncthreads();
    if (threadIdx.x == 0) {
        float sObj = 0.0f, sNoobj = 0.0f, cObj = 0.0f, cNoobj = 0.0f;
        for (int w = 0; w < WAVES_PER_BLK; ++w) {
            sObj   += lds[w * 4 + 0];
            sNoobj += lds[w * 4 + 1];
            cObj   += lds[w * 4 + 2];
            cNoobj += lds[w * 4 + 3];
        }
        const float inv_bs     = 1.0f / (float)N_SAMP;
        const float objScale   = (cObj   > 0.0f) ? inv_bs : 1.0f;
        const float noobjScale = (cNoobj > 0.0f) ? inv_bs : 1.0f;
        out[0] = objScale * sObj + 0.5f * noobjScale * sNoobj;
    }
}

extern "C" void kernel_launch(void* const* d_in, const int* in_sizes, int n_in,
                              void* d_out, int out_size, void* d_ws, size_t ws_size,
                              hipStream_t stream)
{
    (void)in_sizes; (void)n_in; (void)out_size;
    const float* pred = (const float*)d_in[0];
    const float* tgt  = (const float*)d_in[1];
    float* out        = (float*)d_out;
    float* partials   = (float*)d_ws;

    // clamp grid to available scratch (16 bytes of partials per block)
    int nblk = NBLK_MAX;
    size_t max_blk = ws_size / (4 * sizeof(float));
    if ((size_t)nblk > max_blk) nblk = (int)max_blk;
    if (nblk < 1) nblk = 1;

    yolo_loss_main <<<nblk, TPB, 0, stream>>>(pred, tgt, partials);
    yolo_loss_final<<<1,    TPB, 0, stream>>>(partials, nblk, out);
}